// AdaptivePoolAttention_21114059227378
// MI455X (gfx1250) — compile-verified
//
#include <hip/hip_runtime.h>
#include <hip/hip_bf16.h>

typedef float v2f __attribute__((ext_vector_type(2)));
typedef float v4f __attribute__((ext_vector_type(4)));
typedef float v8f __attribute__((ext_vector_type(8)));

#define B_   4
#define T_   64
#define HW_  196
#define D_   768
#define NH_  12
#define HD_  64
#define QKVN (3 * D_)     // 2304
#define NTOK (B_ * T_)    // 256
#define D4_  (D_ / 4)     // 192 float4 per row

// ---------------------------------------------------------------------------
// Kernel 1: spatial mean-pool of x over H*W (the only HBM-heavy op: ~154 MB).
// One block per (b,t) token, 192 threads, one float4 column per thread.
// Streamed with non-temporal b128 loads (read-once data; don't pollute L2).
// ---------------------------------------------------------------------------
__global__ void pool_kernel(const float* __restrict__ x, float* __restrict__ pooled) {
    const int bt  = blockIdx.x;     // 0..255
    const int tid = threadIdx.x;    // 0..191
    const v4f* src = (const v4f*)(x + (size_t)bt * HW_ * D_);
    v4f acc = {};
    for (int hw = 0; hw < HW_; ++hw) {
        v4f val = __builtin_nontemporal_load(src + (size_t)hw * D4_ + tid);
        acc += val;
    }
    const float inv = 1.0f / (float)HW_;
    v4f* dst = (v4f*)(pooled + (size_t)bt * D_);
    dst[tid] = acc * inv;
}

// ---------------------------------------------------------------------------
// Generic fp32 WMMA GEMM: C[M,N] = A[M,K] * B[K,N] (+ bias[N]).
// Row-major A/B/C. One 16x16 tile per wave via V_WMMA_F32_16X16X4_F32.
// Fragment layouts per CDNA5 ISA 7.12.2:
//   A 16x4 : lanes 0-15 -> M=lane, VGPR{0,1}=K{0,1}; lanes 16-31 -> K{2,3}
//   B 4x16 : lanes 0-15 -> N=lane, VGPR{0,1}=K{0,1}; lanes 16-31 -> K{2,3}
//   C 16x16: VGPR r -> M=r (lanes 0-15) / M=r+8 (lanes 16-31), N=lane&15
// Grids are sized so every wave owns a full valid tile (EXEC all-1s for WMMA).
// ---------------------------------------------------------------------------
__global__ void wmma_gemm_kernel(const float* __restrict__ A,
                                 const float* __restrict__ Bm,
                                 const float* __restrict__ bias,
                                 float* __restrict__ C,
                                 int M, int N, int K) {
    const int lane    = threadIdx.x & 31;
    const int wave    = threadIdx.x >> 5;
    const int ntilesN = N >> 4;
    const int tile    = blockIdx.x * (blockDim.x >> 5) + wave;
    const int mt      = tile / ntilesN;
    const int nt      = tile % ntilesN;
    if (mt >= (M >> 4)) return;   // wave-uniform guard

    const int half = lane >> 4;       // selects K pair
    const int l15  = lane & 15;
    const int arow = mt * 16 + l15;   // A row for this lane
    const int bcol = nt * 16 + l15;   // B/C column for this lane

    v8f c = {};
    for (int k0 = 0; k0 < K; k0 += 4) {
        const int kb = k0 + half * 2;
        v2f a = *(const v2f*)(A + (size_t)arow * K + kb);   // contiguous K pair
        v2f b;
        b.x = Bm[(size_t)(kb + 0) * N + bcol];
        b.y = Bm[(size_t)(kb + 1) * N + bcol];
        c = __builtin_amdgcn_wmma_f32_16x16x4_f32(false, a, false, b,
                                                  (short)0, c, false, false);
    }

    const int   rowbase = mt * 16 + half * 8;
    const float bb      = bias ? bias[bcol] : 0.0f;
#pragma unroll
    for (int r = 0; r < 8; ++r) {
        C[(size_t)(rowbase + r) * N + bcol] = c[r] + bb;
    }
}

// ---------------------------------------------------------------------------
// Kernel 3: per-(b,head) attention. Block = 256 threads (8 waves), one block
// per (b,n). Dynamic LDS: q,k,v (LN'd), attn scores, rel_pos_t cache.
//   S = (q*scale) @ k^T          (WMMA, 16 tiles / 8 waves)
//   S += q @ rel_pos_t[dist]^T   (scalar, LDS-resident)
//   P = softmax(S)
//   O = P @ v + q  -> out[b,t,n*64+hd]  (WMMA, fused residual + transpose)
// ---------------------------------------------------------------------------
__global__ void attn_kernel(const float* __restrict__ qkv,
                            const float* __restrict__ gq, const float* __restrict__ bq,
                            const float* __restrict__ gk, const float* __restrict__ bk,
                            const float* __restrict__ gv, const float* __restrict__ bv,
                            const float* __restrict__ relpos,   // (127, 64)
                            float* __restrict__ out) {          // (B,T,768)
    extern __shared__ float smem[];
    float* q    = smem;               // 64*64
    float* kmat = q    + 64 * 64;     // 64*64
    float* v    = kmat + 64 * 64;     // 64*64
    float* attn = v    + 64 * 64;     // 64*64
    float* rel  = attn + 64 * 64;     // 127*64

    const int bn   = blockIdx.x;      // b*NH + n
    const int b    = bn / NH_;
    const int n    = bn % NH_;
    const int tid  = threadIdx.x;
    const int lane = tid & 31;
    const int wave = tid >> 5;

    // cache rel_pos_t in LDS (reused 64x per row)
    for (int i = tid; i < 127 * 64; i += 256) rel[i] = relpos[i];

    // LayerNorm rows of q/k/v into LDS: 192 (matrix,row) jobs
    if (tid < 192) {
        const int m = tid >> 6;       // 0=q, 1=k, 2=v
        const int t = tid & 63;
        const float* src = qkv + (size_t)(b * T_ + t) * QKVN + m * D_ + n * HD_;
        const float* g  = (m == 0) ? gq : ((m == 1) ? gk : gv);
        const float* be = (m == 0) ? bq : ((m == 1) ? bk : bv);
        float* dst = ((m == 0) ? q : ((m == 1) ? kmat : v)) + t * 64;
        float mean = 0.f;
        for (int d = 0; d < 64; ++d) { float xv = src[d]; dst[d] = xv; mean += xv; }
        mean *= (1.0f / 64.0f);
        float var = 0.f;
        for (int d = 0; d < 64; ++d) { float c0 = dst[d] - mean; var += c0 * c0; }
        var *= (1.0f / 64.0f);
        const float rs = rsqrtf(var + 1e-5f);
        for (int d = 0; d < 64; ++d) dst[d] = (dst[d] - mean) * rs * g[d] + be[d];
    }
    __syncthreads();

    const float scale = 0.125f;       // 64^-0.5
    const int   half  = lane >> 4;
    const int   l15   = lane & 15;

    // S = q @ k^T, scaled
    for (int ti = wave; ti < 16; ti += 8) {
        const int mt = ti >> 2, nt = ti & 3;
        v8f c = {};
        for (int k0 = 0; k0 < 64; k0 += 4) {
            const int kb = k0 + half * 2;
            v2f a  = *(const v2f*)(q    + (mt * 16 + l15) * 64 + kb);
            v2f bf = *(const v2f*)(kmat + (nt * 16 + l15) * 64 + kb); // B[k,s]=k[s][k]
            c = __builtin_amdgcn_wmma_f32_16x16x4_f32(false, a, false, bf,
                                                      (short)0, c, false, false);
        }
        const int rowbase = mt * 16 + half * 8;
#pragma unroll
        for (int r = 0; r < 8; ++r)
            attn[(rowbase + r) * 64 + nt * 16 + l15] = c[r] * scale;
    }
    __syncthreads();

    // relative-position bias: attn[t,s] += dot(q[t,:], rel[t-s+63,:])
    {
        const int t  = tid >> 2;
        const int s0 = (tid & 3) * 16;
        const float* qr = q + t * 64;
        for (int s = s0; s < s0 + 16; ++s) {
            const float* rr = rel + (t - s + 63) * 64;
            float acc = 0.f;
            for (int d = 0; d < 64; ++d) acc += qr[d] * rr[d];
            attn[t * 64 + s] += acc;
        }
    }
    __syncthreads();

    // softmax over each row of 64
    if (tid < 64) {
        float* row = attn + tid * 64;
        float mx = row[0];
        for (int s = 1; s < 64; ++s) mx = fmaxf(mx, row[s]);
        float sum = 0.f;
        for (int s = 0; s < 64; ++s) { float e = __expf(row[s] - mx); row[s] = e; sum += e; }
        const float inv = 1.0f / sum;
        for (int s = 0; s < 64; ++s) row[s] *= inv;
    }
    __syncthreads();

    // O = P @ V + q, store transposed into (B,T,NH*HD)
    for (int ti = wave; ti < 16; ti += 8) {
        const int mt = ti >> 2, nt = ti & 3;
        v8f c = {};
        for (int k0 = 0; k0 < 64; k0 += 4) {
            const int kb = k0 + half * 2;
            v2f a = *(const v2f*)(attn + (mt * 16 + l15) * 64 + kb);
            v2f bf;
            bf.x = v[(kb + 0) * 64 + nt * 16 + l15];
            bf.y = v[(kb + 1) * 64 + nt * 16 + l15];
            c = __builtin_amdgcn_wmma_f32_16x16x4_f32(false, a, false, bf,
                                                      (short)0, c, false, false);
        }
        const int rowbase = mt * 16 + half * 8;
        const int col     = nt * 16 + l15;
#pragma unroll
        for (int r = 0; r < 8; ++r) {
            const int t = rowbase + r;
            out[(size_t)(b * T_ + t) * D_ + n * HD_ + col] = c[r] + q[t * 64 + col];
        }
    }
}

// ---------------------------------------------------------------------------
// Host launch
// ---------------------------------------------------------------------------
extern "C" void kernel_launch(void* const* d_in, const int* in_sizes, int n_in,
                              void* d_out, int out_size, void* d_ws, size_t ws_size,
                              hipStream_t stream) {
    const float* x       = (const float*)d_in[0];
    const float* W_qkv   = (const float*)d_in[1];
    const float* g_q     = (const float*)d_in[2];
    const float* b_q     = (const float*)d_in[3];
    const float* g_k     = (const float*)d_in[4];
    const float* b_k     = (const float*)d_in[5];
    const float* g_v     = (const float*)d_in[6];
    const float* b_v     = (const float*)d_in[7];
    const float* W_proj  = (const float*)d_in[8];
    const float* b_proj  = (const float*)d_in[9];
    const float* rel_pos = (const float*)d_in[10];
    float* out = (float*)d_out;

    float* pooled   = (float*)d_ws;                   // 256 x 768
    float* qkv      = pooled   + (size_t)NTOK * D_;   // 256 x 2304
    float* attn_out = qkv      + (size_t)NTOK * QKVN; // 256 x 768

    // 1) pool x over H*W (HBM-bound: ~154 MB, non-temporal b128 stream)
    pool_kernel<<<NTOK, D4_, 0, stream>>>(x, pooled);

    // 2) QKV projection on pooled tokens: (256x768)@(768x2304)
    //    tiles = 16*144 = 2304; 4 waves/block -> 576 blocks (all waves full)
    wmma_gemm_kernel<<<(16 * (QKVN / 16)) / 4, 128, 0, stream>>>(
        pooled, W_qkv, nullptr, qkv, NTOK, QKVN, D_);

    // 3) LN + attention per (b, head)
    const size_t smem_bytes = (size_t)(4 * 64 * 64 + 127 * 64) * sizeof(float); // ~96 KB
    attn_kernel<<<B_ * NH_, 256, smem_bytes, stream>>>(
        qkv, g_q, b_q, g_k, b_k, g_v, b_v, rel_pos, attn_out);

    // 4) output projection + bias: (256x768)@(768x768)
    //    tiles = 16*48 = 768; 4 waves/block -> 192 blocks
    wmma_gemm_kernel<<<(16 * (D_ / 16)) / 4, 128, 0, stream>>>(
        attn_out, W_proj, b_proj, out, NTOK, D_, D_);
}